// HANLayer_2499670966291
// MI455X (gfx1250) — compile-verified
//
#include <hip/hip_runtime.h>
#include <math.h>

typedef float v2f __attribute__((ext_vector_type(2)));
typedef float v8f __attribute__((ext_vector_type(8)));

#define N_NODES 50000
#define N_EDGES 800000
#define IN_DIM  256
#define N_HEADS 8
#define HEAD_D  32
#define FEAT    256
#define HID_DIM 128

__global__ void k_zero(float* __restrict__ p, int n) {
  int i = blockIdx.x * blockDim.x + threadIdx.x;
  int stride = gridDim.x * blockDim.x;
  for (; i < n; i += stride) p[i] = 0.f;
}

// Pack B[K x Nc] into WMMA-fragment order:
// Bp[((tn64*K4 + k4)*4 + c)*64 + lane*2 + p] = B[(k4*4 + koff(lane) + p)*Nc + tn64*64 + c*16 + (lane&15)]
// so a wave reads one fragment as a single coalesced 256B global_load_b64.
__global__ void k_pack_b(const float* __restrict__ B, float* __restrict__ Bp, int K, int Nc) {
  int t = blockIdx.x * blockDim.x + threadIdx.x;
  int K4 = K >> 2;
  int total = (Nc >> 6) * K4 * 4 * 32;
  if (t >= total) return;
  int lane = t & 31;
  int idx  = t >> 5;
  int c    = idx & 3;  idx >>= 2;
  int k4   = idx % K4;
  int tn64 = idx / K4;
  int koff = (lane >> 4) << 1;
  int col  = tn64 * 64 + c * 16 + (lane & 15);
  int kk   = k4 * 4 + koff;
  Bp[(size_t)t * 2 + 0] = B[(size_t)kk * Nc + col];
  Bp[(size_t)t * 2 + 1] = B[(size_t)(kk + 1) * Nc + col];
}

// C[M x Nc] = A[M x K] * Bp(packed), row-major f32, V_WMMA_F32_16X16X4_F32.
// One wave -> 16x64 tile; 8 waves/block share one column-supertile (same packed B stream).
// Double-buffered: next k-step's fragments are in flight while current WMMAs issue.
__global__ void k_wmma_gemm(const float* __restrict__ A, const float* __restrict__ Bp,
                            float* __restrict__ C, int M, int K, int Nc) {
  const int tilesN = Nc >> 6;
  const int bm = blockIdx.x / tilesN;          // 8 row-tiles per block
  const int tn = blockIdx.x - bm * tilesN;
  const int tm = bm * 8 + (threadIdx.x >> 5);
  const int lane = threadIdx.x & 31;
  if (tm * 16 >= M) return;                     // wave-uniform: EXEC all-1 for WMMA
  const int row  = tm * 16 + (lane & 15);
  const int koff = (lane >> 4) << 1;            // lanes 0-15: K=k,k+1 ; lanes 16-31: K=k+2,k+3
  const int col  = tn * 64 + (lane & 15);
  const int K4   = K >> 2;

  const float* Ap = A + (size_t)row * K + koff;
  const float* Bw = Bp + (size_t)tn * K4 * 256 + lane * 2;

  v8f acc0 = {}, acc1 = {}, acc2 = {}, acc3 = {};
  v2f a0 = *(const v2f*)(Ap);
  v2f b0 = *(const v2f*)(Bw);
  v2f b1 = *(const v2f*)(Bw + 64);
  v2f b2 = *(const v2f*)(Bw + 128);
  v2f b3 = *(const v2f*)(Bw + 192);

  for (int k4 = 0; k4 < K4 - 1; ++k4) {
    // prefetch next k-step fragments (loads issue before the waits below retire)
    const float* An = Ap + (k4 + 1) * 4;
    const float* Bn = Bw + (size_t)(k4 + 1) * 256;
    v2f an  = *(const v2f*)(An);
    v2f bn0 = *(const v2f*)(Bn);
    v2f bn1 = *(const v2f*)(Bn + 64);
    v2f bn2 = *(const v2f*)(Bn + 128);
    v2f bn3 = *(const v2f*)(Bn + 192);
    acc0 = __builtin_amdgcn_wmma_f32_16x16x4_f32(false, a0, false, b0, (short)0, acc0, false, false);
    acc1 = __builtin_amdgcn_wmma_f32_16x16x4_f32(false, a0, false, b1, (short)0, acc1, false, false);
    acc2 = __builtin_amdgcn_wmma_f32_16x16x4_f32(false, a0, false, b2, (short)0, acc2, false, false);
    acc3 = __builtin_amdgcn_wmma_f32_16x16x4_f32(false, a0, false, b3, (short)0, acc3, false, false);
    a0 = an; b0 = bn0; b1 = bn1; b2 = bn2; b3 = bn3;
  }
  acc0 = __builtin_amdgcn_wmma_f32_16x16x4_f32(false, a0, false, b0, (short)0, acc0, false, false);
  acc1 = __builtin_amdgcn_wmma_f32_16x16x4_f32(false, a0, false, b1, (short)0, acc1, false, false);
  acc2 = __builtin_amdgcn_wmma_f32_16x16x4_f32(false, a0, false, b2, (short)0, acc2, false, false);
  acc3 = __builtin_amdgcn_wmma_f32_16x16x4_f32(false, a0, false, b3, (short)0, acc3, false, false);

  // C/D layout: VGPR j -> row tm*16 + j (lanes 0-15) or + j+8 (lanes 16-31)
  const int rbase = tm * 16 + ((lane >> 4) << 3);
  #pragma unroll
  for (int j = 0; j < 8; ++j) {
    float* Cr = C + (size_t)(rbase + j) * Nc + col;
    Cr[0]  = acc0[j];
    Cr[16] = acc1[j];
    Cr[32] = acc2[j];
    Cr[48] = acc3[j];
  }
}

// el[n,h] = sum_d z[n,h,d]*al[h,d] ; er likewise
__global__ void k_el_er(const float* __restrict__ z, const float* __restrict__ al,
                        const float* __restrict__ ar, float* __restrict__ el,
                        float* __restrict__ er) {
  int t = blockIdx.x * blockDim.x + threadIdx.x;   // n*8 + h
  if (t >= N_NODES * N_HEADS) return;
  int n = t >> 3, h = t & 7;
  const float* zr  = z + (size_t)n * FEAT + h * HEAD_D;
  const float* alr = al + h * HEAD_D;
  const float* arr = ar + h * HEAD_D;
  float sl = 0.f, sr = 0.f;
  #pragma unroll
  for (int d = 0; d < HEAD_D; ++d) { float v = zr[d]; sl += v * alr[d]; sr += v * arr[d]; }
  el[t] = sl;
  er[t] = sr;
}

// ee[e,h] = exp(leaky_relu(el[src]+er[dst])); denom[dst,h] += ee
// (max-subtraction skipped: logits are O(1), exp is safe; identical softmax result)
__global__ void k_edge_ee(const float* __restrict__ el, const float* __restrict__ er,
                          const int* __restrict__ src, const int* __restrict__ dst,
                          float* __restrict__ ee, float* __restrict__ denom) {
  int e = blockIdx.x * blockDim.x + threadIdx.x;
  if (e >= N_EDGES) return;
  int s = src[e], d = dst[e];
  const float* els = el + (size_t)s * N_HEADS;
  const float* erd = er + (size_t)d * N_HEADS;
  float* eer = ee + (size_t)e * N_HEADS;
  #pragma unroll
  for (int h = 0; h < N_HEADS; ++h) {
    float x = els[h] + erd[h];
    x = x > 0.f ? x : 0.2f * x;
    float ex = expf(x);
    eer[h] = ex;
    atomicAdd(&denom[(size_t)d * N_HEADS + h], ex);
  }
}

// out[dst, :] += alpha[e,h] * z[src, :]  -- one wave per edge, 8 feats per lane.
// z and out are L2-resident (~50 MB each) so the gather+atomic storm stays on-chip.
__global__ void k_msg(const float* __restrict__ z, const float* __restrict__ ee,
                      const float* __restrict__ denom, const int* __restrict__ src,
                      const int* __restrict__ dst, float* __restrict__ out) {
  int wave = (blockIdx.x * blockDim.x + threadIdx.x) >> 5;
  int lane = threadIdx.x & 31;
  if (wave >= N_EDGES) return;
  int s = src[wave], d = dst[wave];
  float a = 0.f;
  if (lane < N_HEADS) {
    float dn = denom[(size_t)d * N_HEADS + lane];
    a = ee[(size_t)wave * N_HEADS + lane] / fmaxf(dn, 1e-9f);
  }
  const float* zs = z + (size_t)s * FEAT;
  float* od = out + (size_t)d * FEAT;
  #pragma unroll
  for (int h = 0; h < N_HEADS; ++h) {
    float ah = __shfl(a, h, 32);
    int f = h * HEAD_D + lane;       // contiguous 128B per wave
    atomicAdd(&od[f], ah * zs[f]);
  }
}

// zf = elu(zf + b), in place
__global__ void k_bias_elu(float* __restrict__ zf, const float* __restrict__ b) {
  long t = (long)blockIdx.x * blockDim.x + threadIdx.x;
  if (t >= (long)N_NODES * FEAT) return;
  float x = zf[t] + b[t & (FEAT - 1)];
  zf[t] = x > 0.f ? x : expm1f(x);
}

// wsum += sum_n ( tanh(t[n,:]+p1_b) . p2_w )  -- one wave per node
__global__ void k_sem_reduce(const float* __restrict__ t, const float* __restrict__ p1b,
                             const float* __restrict__ p2w, float* __restrict__ wsum) {
  int wave = (blockIdx.x * blockDim.x + threadIdx.x) >> 5;
  int lane = threadIdx.x & 31;
  if (wave >= N_NODES) return;
  const float* tr = t + (size_t)wave * HID_DIM;
  float s = 0.f;
  #pragma unroll
  for (int i = 0; i < HID_DIM / 32; ++i) {
    int j = i * 32 + lane;
    s += tanhf(tr[j] + p1b[j]) * p2w[j];
  }
  #pragma unroll
  for (int o = 16; o > 0; o >>= 1) s += __shfl_down(s, o, 32);
  if (lane == 0) atomicAdd(wsum, s);
}

// z0f = beta0*z0f + beta1*z1f  (beta = softmax over the two mean scores)
__global__ void k_final(float* __restrict__ z0f, const float* __restrict__ z1f,
                        const float* __restrict__ wsum) {
  long t = (long)blockIdx.x * blockDim.x + threadIdx.x;
  if (t >= (long)N_NODES * FEAT) return;
  float w0 = wsum[0] * (1.f / N_NODES);
  float w1 = wsum[1] * (1.f / N_NODES);
  float m  = fmaxf(w0, w1);
  float e0 = expf(w0 - m), e1 = expf(w1 - m);
  float inv = 1.f / (e0 + e1);
  z0f[t] = (e0 * inv) * z0f[t] + (e1 * inv) * z1f[t];
}

extern "C" void kernel_launch(void* const* d_in, const int* in_sizes, int n_in,
                              void* d_out, int out_size, void* d_ws, size_t ws_size,
                              hipStream_t stream) {
  (void)in_sizes; (void)n_in; (void)out_size; (void)ws_size;
  const float* h   = (const float*)d_in[0];
  const int* es0   = (const int*)d_in[1];
  const int* ed0   = (const int*)d_in[2];
  const int* es1   = (const int*)d_in[3];
  const int* ed1   = (const int*)d_in[4];
  const float* W0  = (const float*)d_in[5];
  const float* al0 = (const float*)d_in[6];
  const float* ar0 = (const float*)d_in[7];
  const float* b0  = (const float*)d_in[8];
  const float* W1  = (const float*)d_in[9];
  const float* al1 = (const float*)d_in[10];
  const float* ar1 = (const float*)d_in[11];
  const float* b1  = (const float*)d_in[12];
  const float* p1w = (const float*)d_in[13];
  const float* p1b = (const float*)d_in[14];
  const float* p2w = (const float*)d_in[15];
  float* out = (float*)d_out;

  // workspace carve-up (floats): 2*12.8M + 6.4M + 3*0.4M + packs ~= 134 MB
  float* ws     = (float*)d_ws;
  float* zpre   = ws;                                        // N*FEAT (reused per metapath)
  float* z1f    = zpre   + (size_t)N_NODES * FEAT;           // N*FEAT
  float* ebuf   = z1f    + (size_t)N_NODES * FEAT;           // E*8 == N*128 (ee, then t)
  float* el     = ebuf   + (size_t)N_EDGES * N_HEADS;
  float* er     = el     + (size_t)N_NODES * N_HEADS;
  float* denom  = er     + (size_t)N_NODES * N_HEADS;
  float* wsum   = denom  + (size_t)N_NODES * N_HEADS;        // 2 floats (padded to 64)
  float* bpackW = wsum   + 64;                               // IN_DIM*FEAT  (64K floats)
  float* bpackP = bpackW + (size_t)IN_DIM * FEAT;            // FEAT*HID_DIM (32K floats)

  k_zero<<<1, 64, 0, stream>>>(wsum, 2);
  // pack p1_w once (shared by both metapaths)
  {
    int total = (HID_DIM / 64) * (FEAT / 4) * 4 * 32;
    k_pack_b<<<(total + 255) / 256, 256, 0, stream>>>(p1w, bpackP, FEAT, HID_DIM);
  }

  auto metapath = [&](const float* W, const float* al, const float* ar,
                      const float* bb, const int* es, const int* ed,
                      float* zf, float* wacc) {
    // pack W, then z = h @ W via WMMA
    int totalW = (FEAT / 64) * (IN_DIM / 4) * 4 * 32;
    k_pack_b<<<(totalW + 255) / 256, 256, 0, stream>>>(W, bpackW, IN_DIM, FEAT);
    int blocks1 = ((N_NODES / 16 + 7) / 8) * (FEAT / 64);
    k_wmma_gemm<<<blocks1, 256, 0, stream>>>(h, bpackW, zpre, N_NODES, IN_DIM, FEAT);

    k_el_er<<<(N_NODES * N_HEADS + 255) / 256, 256, 0, stream>>>(zpre, al, ar, el, er);
    k_zero<<<1024, 256, 0, stream>>>(denom, N_NODES * N_HEADS);
    k_zero<<<2048, 256, 0, stream>>>(zf, N_NODES * FEAT);
    k_edge_ee<<<(N_EDGES + 255) / 256, 256, 0, stream>>>(el, er, es, ed, ebuf, denom);
    k_msg<<<(N_EDGES * 32 + 255) / 256, 256, 0, stream>>>(zpre, ebuf, denom, es, ed, zf);
    k_bias_elu<<<(N_NODES * FEAT + 255) / 256, 256, 0, stream>>>(zf, bb);

    // t = zf @ p1_w via WMMA, then semantic score accumulation
    int blocks2 = ((N_NODES / 16 + 7) / 8) * (HID_DIM / 64);
    k_wmma_gemm<<<blocks2, 256, 0, stream>>>(zf, bpackP, ebuf, N_NODES, FEAT, HID_DIM);
    k_sem_reduce<<<(N_NODES * 32 + 255) / 256, 256, 0, stream>>>(ebuf, p1b, p2w, wacc);
  };

  metapath(W0, al0, ar0, b0, es0, ed0, out, wsum);      // z0f lives in d_out
  metapath(W1, al1, ar1, b1, es1, ed1, z1f, wsum + 1);
  k_final<<<(int)(((long)N_NODES * FEAT + 255) / 256), 256, 0, stream>>>(out, z1f, wsum);
}